// abstract_fully_connected_17033840296085
// MI455X (gfx1250) — compile-verified
//
#include <hip/hip_runtime.h>
#include <hip/hip_bf16.h>

// SPU abstract-transformer forward (elementwise, FC = 2,097,152).
// Bandwidth-bound streaming kernel: 32 B/element -> 64 MB @ 23.3 TB/s ~ 2.8 us floor.
// 4 elements/thread with B128 loads & stores. The bisection (the only expensive
// compute path) is merged across the 4 elements of a thread into ONE loop with
// per-element freeze flags: wave cost becomes max(iters) instead of sum(iters),
// and each iteration issues 4 independent v_exp_f32 that pipeline in the trans
// unit. Sigmoids use v_rcp_f32 instead of IEEE division (~10x fewer VALU ops
// per sigmoid; ~1 ulp, negligible vs the algorithm's 1e-3 bisection scale).

__device__ __forceinline__ float fast_rcp(float x) {
    return __builtin_amdgcn_rcpf(x);          // v_rcp_f32
}

__device__ __forceinline__ float fast_sigmoid(float x) {
    return fast_rcp(1.0f + __expf(-x));       // v_exp_f32 + v_rcp_f32
}

__device__ __forceinline__ float link_fn(float x) {
    // sin(x)/2 + 0.5 ; |x| <= 0.1 here
    return __sinf(x) * 0.5f + 0.5f;
}

// Everything except the bisection branch.
struct SpuPre {
    float lo, up, usq, l2, l3, sl;
    float lo_out_noC;     // lower_out if caseA/caseB, else lo_C (fully computed here)
    float up_noBisect;    // upper_out if bisection not needed
    bool  needBisect;
};

__device__ __forceinline__ SpuPre spu_pre(float lo, float up, float4 lam) {
    SpuPre r;
    // replicate: upper[upper == lower] += 1e-8
    if (up == lo) up += 1e-8f;
    const float den = up - lo;

    const float l0 = link_fn(lam.x);
    const float l1 = link_fn(lam.y);
    const float l2 = link_fn(lam.z);
    const float l3 = link_fn(lam.w);

    const float sl  = fast_sigmoid(lo);
    const float su  = fast_sigmoid(up);
    const float usq = up * up - 0.5f;

    // ---- Case A: lower >= 0 ----
    const float lamA = lo + den * l0;
    const float lo_A = 2.0f * lamA * lo + (-lamA * lamA - 0.5f);
    const float up_A = usq;

    // ---- Case B: upper <= 0 ----
    const float lo_B = -su;
    const float lamB = lo + den * l1;
    const float sB   = fast_sigmoid(lamB);
    const float wuB  = -sB * (1.0f - sB);
    const float up_B = wuB * lo + (-wuB * lamB - sB);

    // ---- Case C: lower < 0 < upper ----
    const float slope_tangent = -sl * (1.0f - sl);
    const float slope_line    = (usq + sl) * fast_rcp(den);
    const float up_C1         = fmaxf(-sl, usq);

    // C lower branch (no bisection needed)
    const float safe_neg = (lo < 0.0f) ? -lo : 1.0f;
    const float lo_C1 = -0.5f + (1.0f - 2.0f * l2) * ((sl - 0.5f) * fast_rcp(safe_neg)) * up;
    const float lamC2 = up * (2.0f * l2 - 1.0f);
    const float lo_C2 = 2.0f * lamC2 * lo + (-lamC2 * lamC2 - 0.5f);
    const float lo_C  = (l2 < 0.5f) ? lo_C1 : lo_C2;

    const bool caseA = (lo >= 0.0f);
    const bool caseB = (up <= 0.0f);
    const bool caseC = !(caseA || caseB);

    r.lo = lo; r.up = up; r.usq = usq; r.l2 = l2; r.l3 = l3; r.sl = sl;
    r.needBisect   = caseC && (slope_line < slope_tangent);
    r.lo_out_noC   = caseA ? lo_A : (caseB ? lo_B : lo_C);
    r.up_noBisect  = caseA ? up_A : (caseB ? up_B : up_C1);
    return r;
}

__global__ __launch_bounds__(256) void spu_forward_kernel(
    const float*  __restrict__ lower,
    const float*  __restrict__ upper,
    const float4* __restrict__ lambda4,   // one float4 per element (row of lambda_list)
    float*        __restrict__ lower_out,
    float*        __restrict__ upper_out,
    int n)
{
    const int tid = blockIdx.x * blockDim.x + threadIdx.x;
    const int i   = tid * 4;
    if (i >= n) return;

    if (i + 3 < n) {
        // gfx1250 global_prefetch_b8: pull the next tile's lambda rows toward GL2.
        {
            int pf = i + 4 * 256;               // one block-stride ahead
            if (pf + 3 < n) {
                __builtin_prefetch((const void*)(lambda4 + pf), 0, 1);
            }
        }

        // B128 loads (fully coalesced, 16B/lane)
        const float4 lo4 = *(const float4*)(lower + i);
        const float4 up4 = *(const float4*)(upper + i);
        const float4 la[4] = { lambda4[i + 0], lambda4[i + 1],
                               lambda4[i + 2], lambda4[i + 3] };
        const float lo[4] = { lo4.x, lo4.y, lo4.z, lo4.w };
        const float up[4] = { up4.x, up4.y, up4.z, up4.w };

        SpuPre p[4];
#pragma unroll
        for (int j = 0; j < 4; ++j) p[j] = spu_pre(lo[j], up[j], la[j]);

        float lout[4], uout[4];
#pragma unroll
        for (int j = 0; j < 4; ++j) {
            lout[j] = p[j].lo_out_noC;
            uout[j] = p[j].up_noBisect;
        }

        // ---- Merged bisection across the 4 elements (single loop, 4-way ILP) ----
        // Elements that don't need it start converged (st = ed = 0).
        const bool anyNeed = p[0].needBisect | p[1].needBisect |
                             p[2].needBisect | p[3].needBisect;
        if (anyNeed) {
            float st[4], ed[4];
#pragma unroll
            for (int j = 0; j < 4; ++j) {
                st[j] = p[j].needBisect ? p[j].lo : 0.0f;
                ed[j] = 0.0f;
            }
#pragma unroll 1
            for (int it = 0; it < 24; ++it) {
                bool a[4];
                bool any = false;
#pragma unroll
                for (int j = 0; j < 4; ++j) {
                    a[j] = (ed[j] - st[j]) > 1e-3f;
                    any |= a[j];
                }
                if (!any) break;
                // 4 independent sigmoids per iteration -> pipelined v_exp_f32
#pragma unroll
                for (int j = 0; j < 4; ++j) {
                    const float mid  = (st[j] + ed[j]) * 0.5f;
                    const float sm   = fast_sigmoid(mid);
                    const float y_up = (-sm * (1.0f - sm)) * (p[j].up - mid) - sm;
                    const bool right = y_up > p[j].usq;
                    st[j] = (a[j] &&  right) ? mid : st[j];
                    ed[j] = (a[j] && !right) ? mid : ed[j];
                }
            }
#pragma unroll
            for (int j = 0; j < 4; ++j) {
                if (p[j].needBisect) {
                    const float lamC = p[j].lo + (st[j] - p[j].lo) * p[j].l3;
                    const float sC   = fast_sigmoid(lamC);
                    const float wuC  = -sC * (1.0f - sC);
                    uout[j] = wuC * p[j].lo + (-wuC * lamC - sC);
                }
            }
        }

        *(float4*)(lower_out + i) = make_float4(lout[0], lout[1], lout[2], lout[3]);
        *(float4*)(upper_out + i) = make_float4(uout[0], uout[1], uout[2], uout[3]);
    } else {
        // Tail (n not a multiple of 4 — not hit for FC = 2^21, kept for safety)
        for (int k = i; k < n; ++k) {
            SpuPre p = spu_pre(lower[k], upper[k], lambda4[k]);
            float uo = p.up_noBisect;
            if (p.needBisect) {
                float st = p.lo, ed = 0.0f;
#pragma unroll 1
                for (int it = 0; it < 24; ++it) {
                    if (!((ed - st) > 1e-3f)) break;
                    const float mid  = (st + ed) * 0.5f;
                    const float sm   = fast_sigmoid(mid);
                    const float y_up = (-sm * (1.0f - sm)) * (p.up - mid) - sm;
                    if (y_up > p.usq) st = mid; else ed = mid;
                }
                const float lamC = p.lo + (st - p.lo) * p.l3;
                const float sC   = fast_sigmoid(lamC);
                const float wuC  = -sC * (1.0f - sC);
                uo = wuC * p.lo + (-wuC * lamC - sC);
            }
            lower_out[k] = p.lo_out_noC;
            upper_out[k] = uo;
        }
    }
}

extern "C" void kernel_launch(void* const* d_in, const int* in_sizes, int n_in,
                              void* d_out, int out_size, void* d_ws, size_t ws_size,
                              hipStream_t stream) {
    const float*  lower   = (const float*)d_in[0];
    const float*  upper   = (const float*)d_in[1];
    const float4* lambda4 = (const float4*)d_in[2];   // (FC, 4) row-major -> float4 per element
    float* out = (float*)d_out;

    const int n = in_sizes[0];
    float* lower_out = out;          // tuple output, concatenated flat
    float* upper_out = out + n;

    const int threads_needed = (n + 3) / 4;
    const int block = 256;                       // 8 wave32 waves per block
    const int grid  = (threads_needed + block - 1) / block;

    spu_forward_kernel<<<grid, block, 0, stream>>>(lower, upper, lambda4,
                                                   lower_out, upper_out, n);
}